// ONINorm_41592463295064
// MI455X (gfx1250) — compile-verified
//
#include <hip/hip_runtime.h>
#include <hip/hip_bf16.h>

typedef __attribute__((ext_vector_type(2))) float v2f;
typedef __attribute__((ext_vector_type(8))) float v8f;

#define OC 2048
#define FI 18432
#define G  16
#define N  128           // rows per group
#define T_ITERS 5
#define EPS_F 1e-5f

#define KSPLIT 36
#define KCHUNK (FI / KSPLIT)   // 512
#define KB 32                  // staged slab width (cov)
#define KT 64                  // output tile width (final)

// workspace layout (float offsets)
#define WS_S     0
#define WS_B     (WS_S   + G*N*N)
#define WS_T1    (WS_B   + G*N*N)
#define WS_T2    (WS_T1  + G*N*N)
#define WS_RSUM  (WS_T2  + G*N*N)
#define WS_BVEC  (WS_RSUM + OC)
#define WS_SCAL  (WS_BVEC + OC)

#define AS1 __attribute__((address_space(1)))
#define AS3 __attribute__((address_space(3)))

// 16-byte vector type matching the async-load builtin's parameter type
typedef int b128_t __attribute__((vector_size(16)));

__device__ __forceinline__ v8f wmma4(v2f a, v2f b, v8f c) {
    // D = A(16x4 f32) * B(4x16 f32) + C(16x16 f32)
    return __builtin_amdgcn_wmma_f32_16x16x4_f32(
        /*neg_a=*/false, a, /*neg_b=*/false, b,
        /*c_mod=*/(short)0, c, /*reuse_a=*/false, /*reuse_b=*/false);
}

// 16-byte global->LDS copy; async (ASYNCcnt-tracked) on gfx1250 when available.
__device__ __forceinline__ void async_cp16(float* lds_dst, const float* gsrc) {
#if defined(__AMDGCN__) && __has_builtin(__builtin_amdgcn_global_load_async_to_lds_b128)
    __builtin_amdgcn_global_load_async_to_lds_b128(
        (AS1 b128_t*)gsrc, (AS3 b128_t*)lds_dst, 0, 0);
#else
    *(float4*)lds_dst = *(const float4*)gsrc;
#endif
}

__device__ __forceinline__ void wait_async_all() {
#if defined(__AMDGCN__)
    asm volatile("s_wait_asynccnt 0" ::: "memory");
#endif
}

__global__ void zero_kernel(float* __restrict__ p, int n) {
    int i = blockIdx.x * blockDim.x + threadIdx.x;
    if (i < n) p[i] = 0.0f;
}

// -------- Phase 1: S_raw = Z * Z^T (per group), plus row sums --------
__global__ __launch_bounds__(256)
void cov_kernel(const float* __restrict__ Wt, float* __restrict__ S,
                float* __restrict__ rowsum) {
    const int g = blockIdx.y;
    const int chunk = blockIdx.x;
    const int t = threadIdx.x;
    const int w = t >> 5;              // wave id (wave32)
    const int lane = t & 31;
    const int half = lane >> 4;        // 0: K={0,1}, 1: K={2,3}
    const int m = lane & 15;

    __shared__ float Zs[2][N][KB + 4];

    v8f acc[8] = {};
    float rsum = 0.0f;

    const int k0 = chunk * KCHUNK;
    const float* Zg = Wt + (size_t)g * N * FI;

    // prologue: async-stage slab 0
#pragma unroll
    for (int i = 0; i < (N * KB) / (256 * 4); ++i) {
        int linear = i * 256 + t;        // float4 index
        int row = linear >> 3;           // KB/4 = 8 float4 per row
        int c4 = linear & 7;
        async_cp16(&Zs[0][row][c4 * 4], Zg + row * FI + k0 + c4 * 4);
    }
    wait_async_all();
    __syncthreads();

    int cur = 0;
    for (int kb = 0; kb < KCHUNK; kb += KB) {
        // issue async loads for the next slab while we compute this one
        if (kb + KB < KCHUNK) {
            const int nxt = cur ^ 1;
            const int kcol = k0 + kb + KB;
#pragma unroll
            for (int i = 0; i < (N * KB) / (256 * 4); ++i) {
                int linear = i * 256 + t;
                int row = linear >> 3;
                int c4 = linear & 7;
                async_cp16(&Zs[nxt][row][c4 * 4], Zg + row * FI + kcol + c4 * 4);
            }
        }

        if (t < N) {
#pragma unroll
            for (int c = 0; c < KB; ++c) rsum += Zs[cur][t][c];
        }

        const int rowA = 16 * w + m;
#pragma unroll
        for (int kk = 0; kk < KB; kk += 4) {
            v2f a = *(const v2f*)&Zs[cur][rowA][kk + 2 * half];
#pragma unroll
            for (int ct = 0; ct < 8; ++ct) {
                // B = Z^T: B[k][j] = Z[j][k] -> same lane layout as A
                v2f b = *(const v2f*)&Zs[cur][16 * ct + m][kk + 2 * half];
                acc[ct] = wmma4(a, b, acc[ct]);
            }
        }

        wait_async_all();   // next slab fully in LDS
        __syncthreads();    // everyone done reading 'cur' before it is reused
        cur ^= 1;
    }

    float* Sg = S + g * N * N;
#pragma unroll
    for (int ct = 0; ct < 8; ++ct)
#pragma unroll
        for (int r = 0; r < 8; ++r)
            atomicAdd(&Sg[(16 * w + r + 8 * half) * N + 16 * ct + m], acc[ct][r]);
    if (t < N) atomicAdd(&rowsum[g * N + t], rsum);
}

// -------- Phase 2: center-correct, +eps*I, Frobenius normalize, B = I --------
__global__ __launch_bounds__(256)
void prep_kernel(float* __restrict__ S, float* __restrict__ Bm,
                 const float* __restrict__ rowsum, float* __restrict__ scal) {
    const int g = blockIdx.x, t = threadIdx.x;
    __shared__ float rs[N];
    __shared__ float red[256];
    if (t < N) rs[t] = rowsum[g * N + t];
    __syncthreads();

    float* Sg = S + g * N * N;
    float* Bg = Bm + g * N * N;
    const int col = t & (N - 1);
    const int rbase = t >> 7;            // 0 or 1
    const float inv_fi = 1.0f / (float)FI;

    float v[64];
    float ss = 0.0f;
#pragma unroll
    for (int i = 0; i < 64; ++i) {
        int row = 2 * i + rbase;
        float x = Sg[row * N + col] - rs[row] * rs[col] * inv_fi;
        if (row == col) x += EPS_F;
        v[i] = x;
        ss += x * x;
    }
    red[t] = ss;
    __syncthreads();
    for (int s = 128; s > 0; s >>= 1) {
        if (t < s) red[t] += red[t + s];
        __syncthreads();
    }
    float norm = sqrtf(red[0]);
    float inv = 1.0f / norm;
#pragma unroll
    for (int i = 0; i < 64; ++i) {
        int row = 2 * i + rbase;
        Sg[row * N + col] = v[i] * inv;
        Bg[row * N + col] = (row == col) ? 1.0f : 0.0f;
    }
    if (t == 0) scal[g] = rsqrtf(norm);
}

// -------- 128x128x128 fp32 WMMA matmul, C = A * B (global, L2-resident) --------
__device__ __forceinline__
void mm128(float* __restrict__ C, const float* __restrict__ A,
           const float* __restrict__ Bp, int w, int half, int m) {
    v8f acc[8] = {};
    const int rowA = 16 * w + m;
    for (int jj = 0; jj < N; jj += 4) {
        v2f a = *(const v2f*)(A + rowA * N + jj + 2 * half);
#pragma unroll
        for (int ct = 0; ct < 8; ++ct) {
            v2f b;
            b.x = Bp[(jj + 2 * half) * N + 16 * ct + m];
            b.y = Bp[(jj + 2 * half + 1) * N + 16 * ct + m];
            acc[ct] = wmma4(a, b, acc[ct]);
        }
    }
#pragma unroll
    for (int ct = 0; ct < 8; ++ct)
#pragma unroll
        for (int r = 0; r < 8; ++r)
            C[(16 * w + r + 8 * half) * N + 16 * ct + m] = acc[ct][r];
}

// -------- Phase 3: Newton-Schulz, one block per group --------
__global__ __launch_bounds__(256)
void ns_kernel(float* __restrict__ S, float* __restrict__ Bm,
               float* __restrict__ T1, float* __restrict__ T2) {
    const int g = blockIdx.x, t = threadIdx.x;
    const int w = t >> 5, lane = t & 31, half = lane >> 4, m = lane & 15;
    float* Sg  = S  + g * N * N;
    float* Bg  = Bm + g * N * N;
    float* T1g = T1 + g * N * N;
    float* T2g = T2 + g * N * N;

    for (int it = 0; it < T_ITERS; ++it) {
        mm128(T1g, Bg, Bg, w, half, m);      // B^2
        __threadfence(); __syncthreads();
        mm128(T2g, T1g, Bg, w, half, m);     // B^3
        __threadfence(); __syncthreads();

        // acc = B^3 * S ; B = 1.5*B - 0.5*acc
        v8f acc[8] = {};
        const int rowA = 16 * w + m;
        for (int jj = 0; jj < N; jj += 4) {
            v2f a = *(const v2f*)(T2g + rowA * N + jj + 2 * half);
#pragma unroll
            for (int ct = 0; ct < 8; ++ct) {
                v2f b;
                b.x = Sg[(jj + 2 * half) * N + 16 * ct + m];
                b.y = Sg[(jj + 2 * half + 1) * N + 16 * ct + m];
                acc[ct] = wmma4(a, b, acc[ct]);
            }
        }
#pragma unroll
        for (int ct = 0; ct < 8; ++ct)
#pragma unroll
            for (int r = 0; r < 8; ++r) {
                int idx = (16 * w + r + 8 * half) * N + 16 * ct + m;
                Bg[idx] = 1.5f * Bg[idx] - 0.5f * acc[ct][r];
            }
        __threadfence(); __syncthreads();
    }
}

// -------- bvec_i = sum_j B_ij * mean_j --------
__global__ void bvec_kernel(const float* __restrict__ Bm,
                            const float* __restrict__ rowsum,
                            float* __restrict__ bvec) {
    int g = blockIdx.x, i = threadIdx.x;
    const float* Bg = Bm + g * N * N;
    float acc = 0.0f;
    for (int j = 0; j < N; ++j) acc += Bg[i * N + j] * rowsum[g * N + j];
    bvec[g * N + i] = acc / (float)FI;
}

// -------- Phase 4: W = scal * (B*Z - bvec) --------
__global__ __launch_bounds__(256)
void out_kernel(const float* __restrict__ Wt, const float* __restrict__ Bm,
                const float* __restrict__ bvec, const float* __restrict__ scal,
                float* __restrict__ out) {
    const int g = blockIdx.y;
    const int k0 = blockIdx.x * KT;
    const int t = threadIdx.x, w = t >> 5, lane = t & 31;
    const int half = lane >> 4, m = lane & 15;

    __shared__ float Zs[N][KT + 4];
    __shared__ float bv[N];
    if (t < N) bv[t] = bvec[g * N + t];

    const float* Zg = Wt + (size_t)g * N * FI;
#pragma unroll
    for (int i = 0; i < (N * KT) / (256 * 4); ++i) {
        int linear = i * 256 + t;        // float4 index
        int row = linear >> 4;           // KT/4 = 16 float4 per row
        int c4 = linear & 15;
        async_cp16(&Zs[row][c4 * 4], Zg + row * FI + k0 + c4 * 4);
    }
    wait_async_all();
    __syncthreads();

    const float* Bg = Bm + g * N * N;
    v8f acc[4] = {};
    const int rowA = 16 * w + m;
    for (int jj = 0; jj < N; jj += 4) {
        v2f a = *(const v2f*)(Bg + rowA * N + jj + 2 * half);
#pragma unroll
        for (int ct = 0; ct < 4; ++ct) {
            v2f b;
            b.x = Zs[jj + 2 * half][16 * ct + m];
            b.y = Zs[jj + 2 * half + 1][16 * ct + m];
            acc[ct] = wmma4(a, b, acc[ct]);
        }
    }

    float sc = scal[g];
#pragma unroll
    for (int ct = 0; ct < 4; ++ct)
#pragma unroll
        for (int r = 0; r < 8; ++r) {
            int row = 16 * w + r + 8 * half;
            // NT store: output is never re-read; keep the weight resident in L2
            __builtin_nontemporal_store(
                sc * (acc[ct][r] - bv[row]),
                &out[((size_t)(g * N + row)) * FI + k0 + 16 * ct + m]);
        }
}

extern "C" void kernel_launch(void* const* d_in, const int* in_sizes, int n_in,
                              void* d_out, int out_size, void* d_ws, size_t ws_size,
                              hipStream_t stream) {
    const float* Wt = (const float*)d_in[0];
    float* out = (float*)d_out;
    float* ws  = (float*)d_ws;

    float* S      = ws + WS_S;
    float* Bm     = ws + WS_B;
    float* T1     = ws + WS_T1;
    float* T2     = ws + WS_T2;
    float* rowsum = ws + WS_RSUM;
    float* bvec   = ws + WS_BVEC;
    float* scal   = ws + WS_SCAL;

    {
        int nz = G * N * N;
        zero_kernel<<<(nz + 255) / 256, 256, 0, stream>>>(S, nz);
        zero_kernel<<<(OC + 255) / 256, 256, 0, stream>>>(rowsum, OC);
    }

    dim3 gcov(KSPLIT, G);
    cov_kernel<<<gcov, 256, 0, stream>>>(Wt, S, rowsum);

    prep_kernel<<<G, 256, 0, stream>>>(S, Bm, rowsum, scal);

    ns_kernel<<<G, 256, 0, stream>>>(S, Bm, T1, T2);

    bvec_kernel<<<G, N, 0, stream>>>(Bm, rowsum, bvec);

    dim3 gout(FI / KT, G);
    out_kernel<<<gout, 256, 0, stream>>>(Wt, Bm, bvec, scal, out);
}